// SSKernelDiag_28853590294575
// MI455X (gfx1250) — compile-verified
//
#include <hip/hip_runtime.h>
#include <math.h>

// S4D Vandermonde kernel: K[h,l] = 2*Re( sum_n Cmod[h,n] * z[h,n]^l )
// Per-head complex GEMM (16 x 64) x (64 x 128) on V_WMMA_F32_16X16X4_F32
// with l = 16*l1 + l0. B tiles advance across column tiles by a complex
// recurrence (B *= z^256) instead of re-evaluating exponentials, and all
// transcendentals use raw hardware ops (v_exp_f32 = 2^x, v_sin/v_cos in
// revolutions) with constants pre-folded. No divergence -> EXEC all-1s.
// 4 waves per block: wave register demand is ~290 VGPRs (compiler uses
// s_set_vgpr_msb), so keep 1 wave/SIMD to guarantee the workgroup fits
// the WGP register file regardless of its size.

typedef __attribute__((ext_vector_type(2))) float v2f;
typedef __attribute__((ext_vector_type(8))) float v8f;

#define HH 1024
#define NN 64
#define LL 2048
#define WAVES 4          // waves (heads) per block
#define NTILES 8         // 128 columns (l1) / 16

#define LOG2E_F  1.4426950408889634f   // 1/ln(2)
#define INV2PI_F 0.15915494309189535f  // 1/(2*pi)

__device__ __forceinline__ float hexp2(float x)   { return __builtin_amdgcn_exp2f(x); } // 2^x
__device__ __forceinline__ float hsinr(float x)   { return __builtin_amdgcn_sinf(x); }  // sin(2*pi*x)
__device__ __forceinline__ float hcosr(float x)   { return __builtin_amdgcn_cosf(x); }  // cos(2*pi*x)

__global__ __launch_bounds__(32 * WAVES)
void ssd_vandermonde_wmma(const float* __restrict__ log_dt,
                          const float* __restrict__ log_w_real,
                          const float* __restrict__ w_imag,
                          const float* __restrict__ C_re,
                          const float* __restrict__ C_im,
                          float* __restrict__ out)
{
    __shared__ float s_e[WAVES][NN];   // dtA_re * log2(e)   (per-l exponent, base 2)
    __shared__ float s_r[WAVES][NN];   // dtA_im / (2*pi)    (per-l phase, revolutions)
    __shared__ float s_cr[WAVES][NN];  // Re(Cmod)
    __shared__ float s_ci[WAVES][NN];  // Im(Cmod)

    const int lane = threadIdx.x & 31;
    const int wv   = threadIdx.x >> 5;
    const int h    = blockIdx.x * WAVES + wv;

    // ---- Stage 1: per-n scalars (each lane handles n = lane, lane+32) ----
    const float dt = __expf(log_dt[h]);
#pragma unroll
    for (int t = 0; t < 2; ++t) {
        const int n   = lane + 32 * t;
        const int idx = h * NN + n;
        const float wre = -__expf(log_w_real[idx]);
        const float wim = w_imag[idx];
        const float dar = wre * dt;              // Re(dt*A)
        const float dai = wim * dt;              // Im(dt*A)
        // e = exp(dtA) - 1
        const float mg  = __expf(dar);
        const float er  = mg * __cosf(dai) - 1.0f;
        const float ei  = mg * __sinf(dai);
        // tmod = e / w = e * conj(w) / |w|^2
        const float inv = 1.0f / (wre * wre + wim * wim);
        const float tre = (er * wre + ei * wim) * inv;
        const float tim = (ei * wre - er * wim) * inv;
        const float cr  = C_re[idx];
        const float ci  = C_im[idx];
        s_cr[wv][n] = cr * tre - ci * tim;       // Cmod = C * tmod
        s_ci[wv][n] = cr * tim + ci * tre;
        s_e[wv][n]  = dar * LOG2E_F;             // pre-folded hardware units
        s_r[wv][n]  = dai * INV2PI_F;
    }
    __syncthreads();

    // ---- Stage 2: register fragments ----
    // f32 16x4 A layout: lanes 0-15 -> {K=0,K=1}, lanes 16-31 -> {K=2,K=3}; M = lane&15.
    // B 4x16 layout mirrors it: same K pair per lane half, N = lane&15.
    const int   m    = lane & 15;                // A row l0
    const int   koff = (lane < 16) ? 0 : 2;
    const int   ncol = lane & 15;                // N within tile
    const float fm   = (float)m;
    const float tb   = 16.0f * (float)ncol;      // initial 16*l1 (jt = 0)

    v2f aRe[16], aImN[16];                       // A and -Im(A) (F32 WMMA has no A-neg)
    v2f bRe[16], bIm[16];                        // current B tile fragment: z^(16*l1)
    v2f qRe[16], qIm[16];                        // per-column-tile multiplier: z^256

#pragma unroll
    for (int kk = 0; kk < 16; ++kk) {
        const int n0 = kk * 4 + koff;
        const float e0 = s_e[wv][n0], r0 = s_r[wv][n0];
        const float e1 = s_e[wv][n0 + 1], r1 = s_r[wv][n0 + 1];
        const float c0r = s_cr[wv][n0],     c0i = s_ci[wv][n0];
        const float c1r = s_cr[wv][n0 + 1], c1i = s_ci[wv][n0 + 1];

        // A[l0][n] = Cmod[n] * z[n]^l0
        {
            const float g0 = hexp2(e0 * fm), g1 = hexp2(e1 * fm);
            const float p0 = r0 * fm,        p1 = r1 * fm;
            const float v0r = g0 * hcosr(p0), v0i = g0 * hsinr(p0);
            const float v1r = g1 * hcosr(p1), v1i = g1 * hsinr(p1);
            v2f ar, ai;
            ar.x = c0r * v0r - c0i * v0i;
            ar.y = c1r * v1r - c1i * v1i;
            ai.x = -(c0r * v0i + c0i * v0r);
            ai.y = -(c1r * v1i + c1i * v1r);
            aRe[kk]  = ar;
            aImN[kk] = ai;
        }
        // B[n][l1=ncol] = z[n]^(16*ncol)
        {
            const float g0 = hexp2(e0 * tb), g1 = hexp2(e1 * tb);
            const float p0 = r0 * tb,        p1 = r1 * tb;
            v2f br, bi;
            br.x = g0 * hcosr(p0);  br.y = g1 * hcosr(p1);
            bi.x = g0 * hsinr(p0);  bi.y = g1 * hsinr(p1);
            bRe[kk] = br;
            bIm[kk] = bi;
        }
        // q[n] = z[n]^256 (advance one column tile: l1 += 16)
        {
            const float g0 = hexp2(e0 * 256.0f), g1 = hexp2(e1 * 256.0f);
            const float p0 = r0 * 256.0f,        p1 = r1 * 256.0f;
            v2f qr, qi;
            qr.x = g0 * hcosr(p0);  qr.y = g1 * hcosr(p1);
            qi.x = g0 * hsinr(p0);  qi.y = g1 * hsinr(p1);
            qRe[kk] = qr;
            qIm[kk] = qi;
        }
    }

    // ---- Stage 3: 8 column tiles; B advanced by complex recurrence ----
    float* const outh = out + (size_t)h * LL;
    const int rowbase = (lane < 16) ? 0 : 8;

    for (int jt = 0; jt < NTILES; ++jt) {
        v8f acc = {0.f, 0.f, 0.f, 0.f, 0.f, 0.f, 0.f, 0.f};
#pragma unroll
        for (int kk = 0; kk < 16; ++kk) {
            // acc += Re(A)*Re(B);  acc += (-Im(A))*Im(B)  ->  Re(A*B)
            acc = __builtin_amdgcn_wmma_f32_16x16x4_f32(
                      false, aRe[kk],  false, bRe[kk], (short)0, acc, false, false);
            acc = __builtin_amdgcn_wmma_f32_16x16x4_f32(
                      false, aImN[kk], false, bIm[kk], (short)0, acc, false, false);
        }

        // D layout: VGPR r, lanes 0-15 -> M=r, N=lane; lanes 16-31 -> M=r+8.
        const int l1 = jt * 16 + ncol;
#pragma unroll
        for (int r = 0; r < 8; ++r) {
            const int l = l1 * 16 + (r + rowbase);   // l = 16*l1 + l0
            outh[l] = 2.0f * acc[r];
        }

        if (jt < NTILES - 1) {                       // uniform branch (SALU only)
#pragma unroll
            for (int kk = 0; kk < 16; ++kk) {        // B *= z^256
                const v2f br = bRe[kk], bi = bIm[kk];
                const v2f qr = qRe[kk], qi = qIm[kk];
                v2f nr, ni;
                nr.x = br.x * qr.x - bi.x * qi.x;
                nr.y = br.y * qr.y - bi.y * qi.y;
                ni.x = br.x * qi.x + bi.x * qr.x;
                ni.y = br.y * qi.y + bi.y * qr.y;
                bRe[kk] = nr;
                bIm[kk] = ni;
            }
        }
    }
}

extern "C" void kernel_launch(void* const* d_in, const int* in_sizes, int n_in,
                              void* d_out, int out_size, void* d_ws, size_t ws_size,
                              hipStream_t stream) {
    (void)in_sizes; (void)n_in; (void)out_size; (void)d_ws; (void)ws_size;
    const float* log_dt     = (const float*)d_in[0];
    const float* log_w_real = (const float*)d_in[1];
    const float* w_imag     = (const float*)d_in[2];
    const float* C_re       = (const float*)d_in[3];
    const float* C_im       = (const float*)d_in[4];
    // d_in[5] is L (==2048), baked in as a compile-time constant.
    float* out = (float*)d_out;

    dim3 grid(HH / WAVES);   // 256 blocks
    dim3 block(32 * WAVES);  // 128 threads = 4 wave32 (one head per wave)
    ssd_vandermonde_wmma<<<grid, block, 0, stream>>>(
        log_dt, log_w_real, w_imag, C_re, C_im, out);
}